// KVCacheManager_20486994002047
// MI455X (gfx1250) — compile-verified
//
#include <hip/hip_runtime.h>

// KV-cache update + slice. Pure data movement:
//   out[g, s, d] = kv_cache[g, s, d]            for s < seq_len   (256 MB read + 256 MB write)
//   out[g, pos[b], d] = new_kv[g, d]            if pos[b] < seq_len (256 KB, negligible)
// g = ((layer*2+kv)*B + b)*H + h, 512 groups, D=128, L=2048, seq_len=1024.
// Memory-bound: ~512 MB total traffic -> ~22 us at 23.3 TB/s. No FLOPs -> no WMMA.
// CDNA5 path: async global->LDS->global DMA (ASYNCcnt), B128 per lane, contiguous
// 512 B per wave instruction, no VGPR data movement, NT cache policy (dataset
// is 512 MB >> 192 MB L2, each byte touched exactly once).

typedef __attribute__((ext_vector_type(4))) float v4f;
typedef __attribute__((ext_vector_type(4))) int   v4i;

#define AS1 __attribute__((address_space(1)))
#define AS3 __attribute__((address_space(3)))

// gfx12 CPol: TH in bits [2:0]; TH=1 is NT for both loads and stores.
#define CPOL_NT 1

#if __has_builtin(__builtin_amdgcn_global_load_async_to_lds_b128) && \
    __has_builtin(__builtin_amdgcn_global_store_async_from_lds_b128)
#define HAVE_ASYNC_LDS 1
#else
#define HAVE_ASYNC_LDS 0
#endif

static __device__ __forceinline__ void wait_async_0() {
#if __has_builtin(__builtin_amdgcn_s_wait_asynccnt)
  __builtin_amdgcn_s_wait_asynccnt(0);
#else
  asm volatile("s_wait_asynccnt 0" ::: "memory");
#endif
}

// Bulk copy: blockIdx.y = group g (0..511), blockIdx.x tiles the seq_len*32
// float4s of that group's output rows. 256 threads, 8 float4 slots per thread
// -> 32 KB LDS tile per block, each wave's slots private to that wave (so no
// workgroup barrier needed — ASYNCcnt is per-wave and covers exactly its ops).
__global__ void kv_bulk_copy(const float* __restrict__ src, float* __restrict__ dst,
                             unsigned rowQuads /* seq_len*D/4 */,
                             unsigned inRowQuads /* L*D/4 */) {
  const unsigned g    = blockIdx.y;
  const unsigned tid  = threadIdx.x;            // 0..255
  const unsigned base = blockIdx.x * 2048u;     // float4 offset inside group
  const v4f* __restrict__ in4  = ((const v4f*)src) + (size_t)g * inRowQuads;
  v4f* __restrict__       out4 = ((v4f*)dst)      + (size_t)g * rowQuads;

#if HAVE_ASYNC_LDS
  __shared__ v4f lbuf[2048];                    // 32 KB
  if (base + 2048u <= rowQuads) {
    // Full tile: branch-free, 8 back-to-back DMA issues each way.
#pragma unroll
    for (int k = 0; k < 8; ++k) {
      unsigned j = base + (unsigned)k * 256u + tid;
      __builtin_amdgcn_global_load_async_to_lds_b128(
          (AS1 v4i*)(in4 + j), (AS3 v4i*)(lbuf + k * 256 + tid), 0, CPOL_NT);
    }
    wait_async_0();                             // loads done -> LDS tile valid
#pragma unroll
    for (int k = 0; k < 8; ++k) {
      unsigned j = base + (unsigned)k * 256u + tid;
      __builtin_amdgcn_global_store_async_from_lds_b128(
          (AS1 v4i*)(out4 + j), (AS3 v4i*)(lbuf + k * 256 + tid), 0, CPOL_NT);
    }
    wait_async_0();                             // stores drained before LDS freed
  } else {
    // Tail tile (only when seq_len*32 is not a multiple of 2048).
#pragma unroll
    for (int k = 0; k < 8; ++k) {
      unsigned j = base + (unsigned)k * 256u + tid;
      if (j < rowQuads) {
        __builtin_amdgcn_global_load_async_to_lds_b128(
            (AS1 v4i*)(in4 + j), (AS3 v4i*)(lbuf + k * 256 + tid), 0, CPOL_NT);
      }
    }
    wait_async_0();
#pragma unroll
    for (int k = 0; k < 8; ++k) {
      unsigned j = base + (unsigned)k * 256u + tid;
      if (j < rowQuads) {
        __builtin_amdgcn_global_store_async_from_lds_b128(
            (AS1 v4i*)(out4 + j), (AS3 v4i*)(lbuf + k * 256 + tid), 0, CPOL_NT);
      }
    }
    wait_async_0();
  }
#else
  // Fallback: plain wide copy with non-temporal hints.
#pragma unroll
  for (int k = 0; k < 8; ++k) {
    unsigned j = base + (unsigned)k * 256u + tid;
    if (j < rowQuads) {
      v4f v = __builtin_nontemporal_load(in4 + j);
      __builtin_nontemporal_store(v, out4 + j);
    }
  }
#endif
}

// Scatter the single new token into row pos[b] of every group (if visible).
__global__ void kv_scatter_new(const float* __restrict__ new_kv,
                               const int* __restrict__ pos_ids,
                               float* __restrict__ dst,
                               unsigned seq_len) {
  const unsigned g = blockIdx.x;                // 512 groups
  const unsigned d = threadIdx.x;               // 0..127
  const unsigned b = (g >> 3) & 3u;             // g = (((l*2+kv)*4)+b)*8+h
  const int pos = pos_ids[b];                   // Q == 1
  if ((unsigned)pos < seq_len) {
    dst[(size_t)g * seq_len * 128u + (size_t)(unsigned)pos * 128u + d] =
        new_kv[(size_t)g * 128u + d];
  }
}

extern "C" void kernel_launch(void* const* d_in, const int* in_sizes, int n_in,
                              void* d_out, int out_size, void* d_ws, size_t ws_size,
                              hipStream_t stream) {
  (void)n_in; (void)d_ws; (void)ws_size;
  const float* kv_cache = (const float*)d_in[0];
  const float* new_kv   = (const float*)d_in[1];
  const int*   pos_ids  = (const int*)d_in[2];

  const long long GROUPS = 512;  // 8 layers * 2 * B(4) * H(8)
  const long long D      = 128;

  const unsigned L       = (unsigned)((long long)in_sizes[0] / (GROUPS * D)); // 2048
  const unsigned seq_len = (unsigned)((long long)out_size    / (GROUPS * D)); // 1024

  const unsigned rowQuads   = seq_len * (unsigned)(D / 4);  // 32768
  const unsigned inRowQuads = L       * (unsigned)(D / 4);  // 65536

  dim3 grid((rowQuads + 2047u) / 2048u, (unsigned)GROUPS);  // (16, 512)
  kv_bulk_copy<<<grid, 256, 0, stream>>>(kv_cache, (float*)d_out, rowQuads, inRowQuads);
  kv_scatter_new<<<(unsigned)GROUPS, (unsigned)D, 0, stream>>>(new_kv, pos_ids,
                                                               (float*)d_out, seq_len);
}